// PerformerAttentionBlock_62457414419000
// MI455X (gfx1250) — compile-verified
//
#include <hip/hip_runtime.h>
#include <hip/hip_bf16.h>

typedef __attribute__((ext_vector_type(2))) float v2f;
typedef __attribute__((ext_vector_type(8))) float v8f;

#define WMMA_F32(a, b, c) \
  __builtin_amdgcn_wmma_f32_16x16x4_f32(false, (a), false, (b), (short)0, (c), false, false)

#define NEG_INF (-3.402823466e38f)
#define DN    0.3535533906f   /* 64^-0.25 */
#define DN2H  0.0625f         /* 0.5 * 64^-0.5 */
#define RATIO 0.06131393f     /* 266^-0.5 */
#define KEPS  1e-4f

// ---------------------------------------------------------------------------
// LayerNorm: one row (512) per 256-thread block
// ---------------------------------------------------------------------------
__launch_bounds__(256)
__global__ void ln_kernel(const float* __restrict__ x, const float* __restrict__ g,
                          const float* __restrict__ bta, float* __restrict__ out)
{
  __shared__ float red[256];
  const int tid = threadIdx.x;
  const size_t base = (size_t)blockIdx.x * 512;
  float a0 = x[base + tid], a1 = x[base + tid + 256];
  red[tid] = a0 + a1;
  __syncthreads();
  for (int s = 128; s > 0; s >>= 1) { if (tid < s) red[tid] += red[tid + s]; __syncthreads(); }
  float mu = red[0] * (1.0f / 512.0f);
  __syncthreads();
  float d0 = a0 - mu, d1 = a1 - mu;
  red[tid] = d0 * d0 + d1 * d1;
  __syncthreads();
  for (int s = 128; s > 0; s >>= 1) { if (tid < s) red[tid] += red[tid + s]; __syncthreads(); }
  float rstd = rsqrtf(red[0] * (1.0f / 512.0f) + 1e-5f);
  out[base + tid]       = d0 * rstd * g[tid]       + bta[tid];
  out[base + tid + 256] = d1 * rstd * g[tid + 256] + bta[tid + 256];
}

// ---------------------------------------------------------------------------
// Generic fp32 WMMA GEMM: out = epilogue(A[M,K] @ B[K,N] + bias (+res))
// block 256 thr = 8 waves; tile 128x64; each wave 32x32 (2x2 WMMA tiles)
// A/B tiles staged via GLOBAL_LOAD_ASYNC_TO_LDS_B128 (ASYNCcnt path),
// next K-chunk prefetched with global_prefetch_b8.
// As: row-major [128][20] (80B rows, 16B aligned, conflict-free frag reads)
// Bs: [16][72]   (2*72 = 16 mod 64 -> disjoint banks for the two k-halves)
// ---------------------------------------------------------------------------
__launch_bounds__(256)
__global__ void gemm_kernel(const float* __restrict__ A, const float* __restrict__ Bm,
                            const float* __restrict__ bias, const float* __restrict__ res,
                            float* __restrict__ out, int M, int N, int K, int gelu)
{
  __shared__ float As[128 * 20];
  __shared__ float Bs[16 * 72];
  const int tid = threadIdx.x, lane = tid & 31, wv = tid >> 5;
  const int wm = wv & 3, wn = wv >> 2;
  const int m0 = blockIdx.y * 128, n0 = blockIdx.x * 64;
  const int ml = lane & 15;
  const int kk = (lane >> 4) << 1;
  v8f acc[2][2] = {};

  // per-thread async-copy coordinates
  const int arow0 = tid >> 2,          acg = (tid & 3) << 2;   // A part 1: rows 0..63
  const int arow1 = (tid + 256) >> 2;                           // A part 2: rows 64..127
  const int bkr = tid >> 4,            bc  = (tid & 15) << 2;   // B: 16x64

  for (int k0 = 0; k0 < K; k0 += 16) {
    __syncthreads();
    { // async A tile: 128 rows x 16 k, each lane copies 16B
      const float* gp0 = A + (size_t)(m0 + arow0) * K + k0 + acg;
      unsigned l0 = (unsigned)(size_t)&As[arow0 * 20 + acg];
      asm volatile("global_load_async_to_lds_b128 %0, %1, off"
                   :: "v"(l0), "v"(gp0) : "memory");
      const float* gp1 = A + (size_t)(m0 + arow1) * K + k0 + acg;
      unsigned l1 = (unsigned)(size_t)&As[arow1 * 20 + acg];
      asm volatile("global_load_async_to_lds_b128 %0, %1, off"
                   :: "v"(l1), "v"(gp1) : "memory");
    }
    { // async B tile: 16 k x 64 n
      const float* gp = Bm + (size_t)(k0 + bkr) * N + n0 + bc;
      unsigned lb = (unsigned)(size_t)&Bs[bkr * 72 + bc];
      asm volatile("global_load_async_to_lds_b128 %0, %1, off"
                   :: "v"(lb), "v"(gp) : "memory");
    }
    if (k0 + 16 < K) { // prefetch next chunk
      __builtin_prefetch(A + (size_t)(m0 + arow0) * K + k0 + 16 + acg, 0, 1);
      __builtin_prefetch(Bm + (size_t)(k0 + 16 + bkr) * N + n0 + bc, 0, 1);
    }
    asm volatile("s_wait_asynccnt 0" ::: "memory");
    __syncthreads();

    #pragma unroll
    for (int ks = 0; ks < 16; ks += 4) {
      int kl = ks + kk;
      v2f af[2], bf[2];
      #pragma unroll
      for (int i = 0; i < 2; ++i) {
        int mr = wm * 32 + i * 16 + ml;
        af[i].x = As[mr * 20 + kl];   af[i].y = As[mr * 20 + kl + 1];
        int nc = wn * 32 + i * 16 + ml;
        bf[i].x = Bs[kl * 72 + nc];   bf[i].y = Bs[(kl + 1) * 72 + nc];
      }
      #pragma unroll
      for (int i = 0; i < 2; ++i)
        #pragma unroll
        for (int j = 0; j < 2; ++j)
          acc[i][j] = WMMA_F32(af[i], bf[j], acc[i][j]);
    }
  }

  #pragma unroll
  for (int j = 0; j < 2; ++j) {
    int gc = n0 + wn * 32 + j * 16 + ml;
    float bv = bias ? bias[gc] : 0.0f;
    #pragma unroll
    for (int i = 0; i < 2; ++i) {
      #pragma unroll
      for (int r = 0; r < 8; ++r) {
        int gr = m0 + wm * 32 + i * 16 + r + ((lane >> 4) << 3);
        float val = acc[i][j][r] + bv;
        if (gelu) val = 0.5f * val * (1.0f + erff(val * 0.70710678f));
        if (res) val += res[(size_t)gr * N + gc];
        out[(size_t)gr * N + gc] = val;
      }
    }
  }
}

// ---------------------------------------------------------------------------
// diag[bh][n] = 0.5 * dn^2 * sum_d q[b,n,h,d]^2
// ---------------------------------------------------------------------------
__launch_bounds__(256)
__global__ void diag_kernel(const float* __restrict__ qk, float* __restrict__ dg)
{
  int idx = blockIdx.x * 256 + threadIdx.x;
  if (idx >= 64 * 4096) return;
  int bh = idx >> 12, n = idx & 4095;
  int b = bh >> 3, hh = bh & 7;
  const float* p = qk + ((size_t)b * 4096 + n) * 512 + hh * 64;
  float s = 0.0f;
  #pragma unroll
  for (int i = 0; i < 16; ++i) {
    float4 v = *(const float4*)(p + i * 4);
    s += v.x * v.x + v.y * v.y + v.z * v.z + v.w * v.w;
  }
  dg[idx] = s * DN2H;
}

// ---------------------------------------------------------------------------
// Feature map: dd = (q*dn) @ proj^T  [per bh: 4096x64 @ 64x272]
// is_query: row-max + exp in-register -> qp.  else: raw dd + block max.
// block 128 thr = 4 waves, each wave a 16-row tile; 17 col tiles in registers
// ---------------------------------------------------------------------------
__launch_bounds__(128)
__global__ void feat_kernel(const float* __restrict__ qk, const float* __restrict__ proj,
                            const float* __restrict__ diag, float* __restrict__ outp,
                            float* __restrict__ bmax, int is_query)
{
  __shared__ float pT[16 * 273];   // proj^T chunk: pT[k][mf]
  __shared__ float aT[16 * 68];    // A chunk transposed: aT[k][row]
  __shared__ float wred[4];
  const int tid = threadIdx.x, lane = tid & 31, wv = tid >> 5;
  const int bh = blockIdx.y, b = bh >> 3, hh = bh & 7;
  const int nblk = blockIdx.x * 64;
  const int ml = lane & 15;
  const int kk = (lane >> 4) << 1;
  v8f acc[17] = {};

  for (int kc = 0; kc < 4; ++kc) {
    __syncthreads();
    for (int e = tid; e < 272 * 16; e += 128) {
      int mf = e >> 4, kq = e & 15;
      pT[kq * 273 + mf] = (mf < 266) ? proj[mf * 64 + kc * 16 + kq] : 0.0f;
    }
    {
      int row = tid >> 1, kq = (tid & 1) * 8;
      const float* gp = qk + ((size_t)(b * 4096 + nblk + row)) * 512 + hh * 64 + kc * 16 + kq;
      float4 v0 = *(const float4*)gp;
      float4 v1 = *(const float4*)(gp + 4);
      aT[(kq + 0) * 68 + row] = v0.x * DN; aT[(kq + 1) * 68 + row] = v0.y * DN;
      aT[(kq + 2) * 68 + row] = v0.z * DN; aT[(kq + 3) * 68 + row] = v0.w * DN;
      aT[(kq + 4) * 68 + row] = v1.x * DN; aT[(kq + 5) * 68 + row] = v1.y * DN;
      aT[(kq + 6) * 68 + row] = v1.z * DN; aT[(kq + 7) * 68 + row] = v1.w * DN;
    }
    __syncthreads();
    #pragma unroll
    for (int ks = 0; ks < 16; ks += 4) {
      int kl = ks + kk;
      v2f af;
      af.x = aT[kl * 68 + wv * 16 + ml];
      af.y = aT[(kl + 1) * 68 + wv * 16 + ml];
      #pragma unroll
      for (int t = 0; t < 17; ++t) {
        v2f bf;
        bf.x = pT[kl * 273 + t * 16 + ml];
        bf.y = pT[(kl + 1) * 273 + t * 16 + ml];
        acc[t] = WMMA_F32(af, bf, acc[t]);
      }
    }
  }

  if (is_query) {
    #pragma unroll
    for (int r = 0; r < 8; ++r) {
      float mx = NEG_INF;
      #pragma unroll
      for (int t = 0; t < 17; ++t) {
        float cv = acc[t][r];
        if (t == 16 && ml >= 10) cv = NEG_INF;   // mask padded cols 266..271
        mx = fmaxf(mx, cv);
      }
      mx = fmaxf(mx, __shfl_xor(mx, 1, 32));
      mx = fmaxf(mx, __shfl_xor(mx, 2, 32));
      mx = fmaxf(mx, __shfl_xor(mx, 4, 32));
      mx = fmaxf(mx, __shfl_xor(mx, 8, 32));
      int row = nblk + wv * 16 + r + ((lane >> 4) << 3);
      float dv = diag[(size_t)bh * 4096 + row];
      size_t rb = ((size_t)bh * 4096 + row) * 272;
      #pragma unroll
      for (int t = 0; t < 17; ++t) {
        int col = t * 16 + ml;
        float val = (col < 266) ? RATIO * (expf(acc[t][r] - dv - mx) + KEPS) : 0.0f;
        outp[rb + col] = val;
      }
    }
  } else {
    float wmax = NEG_INF;
    #pragma unroll
    for (int r = 0; r < 8; ++r) {
      int row = nblk + wv * 16 + r + ((lane >> 4) << 3);
      size_t rb = ((size_t)bh * 4096 + row) * 272;
      #pragma unroll
      for (int t = 0; t < 17; ++t) {
        int col = t * 16 + ml;
        float cv = acc[t][r];
        outp[rb + col] = cv;
        if (col < 266) wmax = fmaxf(wmax, cv);
      }
    }
    wmax = fmaxf(wmax, __shfl_xor(wmax, 1, 32));
    wmax = fmaxf(wmax, __shfl_xor(wmax, 2, 32));
    wmax = fmaxf(wmax, __shfl_xor(wmax, 4, 32));
    wmax = fmaxf(wmax, __shfl_xor(wmax, 8, 32));
    wmax = fmaxf(wmax, __shfl_xor(wmax, 16, 32));
    if (lane == 0) wred[wv] = wmax;
    __syncthreads();
    if (tid == 0)
      bmax[blockIdx.y * gridDim.x + blockIdx.x] =
        fmaxf(fmaxf(wred[0], wred[1]), fmaxf(wred[2], wred[3]));
  }
}

// ---------------------------------------------------------------------------
// global max over per-block key maxima
// ---------------------------------------------------------------------------
__launch_bounds__(256)
__global__ void maxreduce_kernel(const float* __restrict__ in, int n, float* __restrict__ out)
{
  __shared__ float red[256];
  float m = NEG_INF;
  for (int i = threadIdx.x; i < n; i += 256) m = fmaxf(m, in[i]);
  red[threadIdx.x] = m;
  __syncthreads();
  for (int s = 128; s > 0; s >>= 1) {
    if (threadIdx.x < s) red[threadIdx.x] = fmaxf(red[threadIdx.x], red[threadIdx.x + s]);
    __syncthreads();
  }
  if (threadIdx.x == 0) out[0] = red[0];
}

// ---------------------------------------------------------------------------
// key features pass 2: kp = ratio*(exp(dd - diag - gmax)+eps), zero padding
// ---------------------------------------------------------------------------
__launch_bounds__(256)
__global__ void kexp_kernel(float* __restrict__ kp, const float* __restrict__ dg,
                            const float* __restrict__ gmax, int total)
{
  int idx = blockIdx.x * 256 + threadIdx.x;
  if (idx >= total) return;
  int col = idx % 272;
  int nbh = idx / 272;
  float dd = kp[idx];
  kp[idx] = (col < 266) ? RATIO * (expf(dd - dg[nbh] - gmax[0]) + KEPS) : 0.0f;
}

// ---------------------------------------------------------------------------
// ctx_aug[bh] = kp^T (272x4096) @ [v | 1 | 0] (4096x80); K split over 4 waves,
// partials merged with LDS float atomics; col 64 = sum_n kp
// ---------------------------------------------------------------------------
__launch_bounds__(128)
__global__ void ctx_kernel(const float* __restrict__ kp, const float* __restrict__ v,
                           float* __restrict__ ctx)
{
  __shared__ float cbuf[16 * 80];
  const int tid = threadIdx.x, lane = tid & 31, wv = tid >> 5;
  const int bh = blockIdx.y, b = bh >> 3, hh = bh & 7;
  const int mt = blockIdx.x;
  for (int i = tid; i < 1280; i += 128) cbuf[i] = 0.0f;
  __syncthreads();
  const int ml = lane & 15;
  const int kk = (lane >> 4) << 1;
  v8f acc[5] = {};
  const size_t kbase = (size_t)bh * 4096 * 272;
  const size_t vbase = (size_t)b * 4096 * 512 + hh * 64;
  for (int it = 0; it < 256; ++it) {
    int n = wv * 1024 + it * 4 + kk;
    v2f af;
    af.x = kp[kbase + (size_t)n * 272 + mt * 16 + ml];
    af.y = kp[kbase + (size_t)(n + 1) * 272 + mt * 16 + ml];
    #pragma unroll
    for (int t = 0; t < 5; ++t) {
      int d = t * 16 + ml;
      v2f bf;
      if (t < 4) {
        bf.x = v[vbase + (size_t)n * 512 + d];
        bf.y = v[vbase + (size_t)(n + 1) * 512 + d];
      } else {
        float c = (ml == 0) ? 1.0f : 0.0f;  // d==64 -> ones column
        bf.x = c; bf.y = c;
      }
      acc[t] = WMMA_F32(af, bf, acc[t]);
    }
  }
  #pragma unroll
  for (int t = 0; t < 5; ++t)
    #pragma unroll
    for (int r = 0; r < 8; ++r)
      atomicAdd(&cbuf[(r + ((lane >> 4) << 3)) * 80 + t * 16 + ml], acc[t][r]);
  __syncthreads();
  const size_t cb = (size_t)bh * 272 * 80 + (size_t)mt * 16 * 80;
  for (int i = tid; i < 1280; i += 128) ctx[cb + i] = cbuf[i];
}

// ---------------------------------------------------------------------------
// attn = (qp @ ctx_aug) normalized by column 64; written head-interleaved
// ---------------------------------------------------------------------------
__launch_bounds__(128)
__global__ void attn_kernel(const float* __restrict__ qp, const float* __restrict__ ctx,
                            float* __restrict__ att)
{
  const int tid = threadIdx.x, lane = tid & 31, wv = tid >> 5;
  const int bh = blockIdx.y, b = bh >> 3, hh = bh & 7;
  const int n0 = blockIdx.x * 64 + wv * 16;
  const int ml = lane & 15, kk = (lane >> 4) << 1;
  v8f acc[5] = {};
  const size_t qb = ((size_t)bh * 4096 + n0 + ml) * 272;
  const size_t cb = (size_t)bh * 272 * 80;
  for (int ks = 0; ks < 272; ks += 4) {
    int kl = ks + kk;
    v2f af;
    af.x = qp[qb + kl];
    af.y = qp[qb + kl + 1];
    #pragma unroll
    for (int t = 0; t < 5; ++t) {
      v2f bf;
      bf.x = ctx[cb + (size_t)kl * 80 + t * 16 + ml];
      bf.y = ctx[cb + (size_t)(kl + 1) * 80 + t * 16 + ml];
      acc[t] = WMMA_F32(af, bf, acc[t]);
    }
  }
  #pragma unroll
  for (int r = 0; r < 8; ++r) {
    float dlo = __shfl(acc[4][r], 0, 32);
    float dhi = __shfl(acc[4][r], 16, 32);
    float dinv = 1.0f / ((lane < 16) ? dlo : dhi);
    int n = n0 + r + ((lane >> 4) << 3);
    size_t ob = ((size_t)b * 4096 + n) * 512 + hh * 64;
    #pragma unroll
    for (int t = 0; t < 4; ++t)
      att[ob + t * 16 + ml] = acc[t][r] * dinv;
  }
}

// ---------------------------------------------------------------------------
extern "C" void kernel_launch(void* const* d_in, const int* in_sizes, int n_in,
                              void* d_out, int out_size, void* d_ws, size_t ws_size,
                              hipStream_t stream)
{
  const float* x     = (const float*)d_in[0];
  const float* w_q   = (const float*)d_in[1];
  const float* w_k   = (const float*)d_in[2];
  const float* w_v   = (const float*)d_in[3];
  const float* w_o   = (const float*)d_in[4];
  const float* b_o   = (const float*)d_in[5];
  const float* ln1_g = (const float*)d_in[6];
  const float* ln1_b = (const float*)d_in[7];
  const float* ln2_g = (const float*)d_in[8];
  const float* ln2_b = (const float*)d_in[9];
  const float* ff_w1 = (const float*)d_in[10];
  const float* ff_b1 = (const float*)d_in[11];
  const float* ff_w2 = (const float*)d_in[12];
  const float* ff_b2 = (const float*)d_in[13];
  const float* proj  = (const float*)d_in[14];

  const size_t BN = 32768;           // B*N
  const size_t ACT = BN * 512;       // 16,777,216
  float* h   = (float*)d_ws;
  float* q   = h   + ACT;
  float* k   = q   + ACT;
  float* v   = k   + ACT;
  float* y1  = v   + ACT;
  float* h2  = y1  + ACT;
  float* att = h2  + ACT;
  float* ffm = att + ACT;            // 32768*2048
  float* qp  = ffm + BN * 2048;      // 64*4096*272
  float* kp  = qp  + (size_t)64 * 4096 * 272;
  float* dq  = kp  + (size_t)64 * 4096 * 272;   // 262144
  float* dk  = dq  + 262144;
  float* ctx = dk  + 262144;         // 64*272*80
  float* bm  = ctx + (size_t)64 * 272 * 80;     // 4096
  float* gm  = bm  + 4096;           // 1
  float* out = (float*)d_out;

  // attention sublayer
  ln_kernel<<<32768, 256, 0, stream>>>(x, ln1_g, ln1_b, h);
  gemm_kernel<<<dim3(8, 256), 256, 0, stream>>>(h, w_q, nullptr, nullptr, q, 32768, 512, 512, 0);
  gemm_kernel<<<dim3(8, 256), 256, 0, stream>>>(h, w_k, nullptr, nullptr, k, 32768, 512, 512, 0);
  gemm_kernel<<<dim3(8, 256), 256, 0, stream>>>(h, w_v, nullptr, nullptr, v, 32768, 512, 512, 0);
  diag_kernel<<<1024, 256, 0, stream>>>(q, dq);
  diag_kernel<<<1024, 256, 0, stream>>>(k, dk);
  feat_kernel<<<dim3(64, 64), 128, 0, stream>>>(q, proj, dq, qp, nullptr, 1);
  feat_kernel<<<dim3(64, 64), 128, 0, stream>>>(k, proj, dk, kp, bm, 0);
  maxreduce_kernel<<<1, 256, 0, stream>>>(bm, 4096, gm);
  {
    int total = 64 * 4096 * 272;
    kexp_kernel<<<(total + 255) / 256, 256, 0, stream>>>(kp, dk, gm, total);
  }
  ctx_kernel<<<dim3(17, 64), 128, 0, stream>>>(kp, v, ctx);
  attn_kernel<<<dim3(64, 64), 128, 0, stream>>>(qp, ctx, att);
  gemm_kernel<<<dim3(8, 256), 256, 0, stream>>>(att, w_o, b_o, x, y1, 32768, 512, 512, 0);

  // feed-forward sublayer
  ln_kernel<<<32768, 256, 0, stream>>>(y1, ln2_g, ln2_b, h2);
  gemm_kernel<<<dim3(32, 256), 256, 0, stream>>>(h2, ff_w1, ff_b1, nullptr, ffm, 32768, 2048, 512, 1);
  gemm_kernel<<<dim3(8, 256), 256, 0, stream>>>(ffm, ff_w2, ff_b2, y1, out, 32768, 512, 2048, 0);
}